// LayerGarNet_88321707475161
// MI455X (gfx1250) — compile-verified
//
#include <hip/hip_runtime.h>
#include <math.h>

typedef __attribute__((ext_vector_type(2))) float v2f;
typedef __attribute__((ext_vector_type(8))) float v8f;

#define BATCH 32
#define NV    2048
#define NP    64
#define NA    8
#define NFO   128      // NF output
#define NFEAT 72       // P + A
#define TWOF  144      // 2*F
#define XS_LD 68       // padded LDS row stride for 64-wide X tile:
                       //  bank = (4m+kb)%64 -> conflict-free, 16B-aligned float4 staging
#define AS_LD 10       // padded stride for 8-wide agg tile

__device__ __forceinline__ v8f wmma_f32_k4(v2f a, v2f b, v8f c) {
  // D = A(16x4 f32) * B(4x16 f32) + C(16x16 f32)
  return __builtin_amdgcn_wmma_f32_16x16x4_f32(false, a, false, b, (short)0, c, false, false);
}

// ---------------- Kernel 1: vi = X@W1+b1 ; agg = exp(-|X@W2+b2|) --------------
// 256 blocks x 16 row-tiles. B fragments persistent in VGPRs; X tile staged in
// LDS once per tile and shared by all 5 compute waves.
__global__ __launch_bounds__(160) void
k_feats(const float* __restrict__ X, const float* __restrict__ W1,
        const float* __restrict__ b1, const float* __restrict__ W2,
        const float* __restrict__ b2, float* __restrict__ vi,
        float* __restrict__ agg)
{
  __shared__ float xs[16 * XS_LD];
  const int tid  = threadIdx.x;
  const int lane = tid & 31;
  const int wave = tid >> 5;                 // 0..3: vi n-tiles, 4: agg
  const int m    = lane & 15;
  const int kh   = (lane >> 4) * 2;          // K sub-offset 0 or 2
  const int n    = lane & 15;
  const int blockrow = blockIdx.x * 256;

  // persistent B fragments (reused across all 16 row tiles)
  v2f bfr[16];
  if (wave < 4) {
    const int ncol = wave * 16 + n;
#pragma unroll
    for (int kk = 0; kk < 16; ++kk) {
      const int kb = kk * 4 + kh;
      bfr[kk].x = W1[kb * NP + ncol];
      bfr[kk].y = W1[(kb + 1) * NP + ncol];
    }
  } else {
    const int nn   = (n < NA) ? n : 0;       // clamp index, mask value:
    const float mk = (n < NA) ? 1.0f : 0.0f; // keeps EXEC all-ones around WMMA
#pragma unroll
    for (int kk = 0; kk < 16; ++kk) {
      const int kb = kk * 4 + kh;
      bfr[kk].x = W2[kb * NA + nn] * mk;
      bfr[kk].y = W2[(kb + 1) * NA + nn] * mk;
    }
  }

#pragma unroll 1
  for (int t = 0; t < 16; ++t) {
    const int rowb = blockrow + t * 16;
    // cooperative stage of 16x64 X tile (256 float4 chunks over 160 threads)
    for (int i = tid; i < 256; i += 160) {
      const int r = i >> 4, c = (i & 15) * 4;
      *(float4*)(xs + r * XS_LD + c) =
          *(const float4*)(X + (size_t)(rowb + r) * NP + c);
    }
    __syncthreads();

    v8f acc = {0.f,0.f,0.f,0.f,0.f,0.f,0.f,0.f};
#pragma unroll
    for (int kk = 0; kk < 16; ++kk) {
      const int kb = kk * 4 + kh;
      v2f av; av.x = xs[m * XS_LD + kb]; av.y = xs[m * XS_LD + kb + 1];
      acc = wmma_f32_k4(av, bfr[kk], acc);
    }

    const int mrow = rowb + (lane >> 4) * 8;
    if (wave < 4) {
      const int ncol = wave * 16 + n;
      const float bb = b1[ncol];
#pragma unroll
      for (int r = 0; r < 8; ++r)
        vi[(size_t)(mrow + r) * NP + ncol] = acc[r] + bb;
    } else if (n < NA) {
      const float bb = b2[n];
#pragma unroll
      for (int r = 0; r < 8; ++r)
        agg[(size_t)(mrow + r) * NA + n] = expf(-fabsf(acc[r] + bb));
    }
    __syncthreads();
  }
}

// ------------- Kernel 2: collapsed[b,a,:] = [max_v, mean_v] agg*feats --------
__global__ __launch_bounds__(96) void
k_collapse(const float* __restrict__ vi, const float* __restrict__ agg,
           float* __restrict__ collapsed)
{
  const int a = blockIdx.x;                      // 0..7
  const int b = blockIdx.y;                      // 0..31
  const int f = threadIdx.x;                     // feature this thread owns
  if (f >= NFEAT) return;
  const float* aggB = agg + (size_t)b * NV * NA;
  const float* viB  = vi  + (size_t)b * NV * NP;
  float mx = -__builtin_inff();
  float sm = 0.0f;
  if (f < NP) {
#pragma unroll 4
    for (int v = 0; v < NV; ++v) {
      float e = aggB[v * NA + a];
      float p = e * viB[v * NP + f];
      mx = fmaxf(mx, p); sm += p;
    }
  } else {
    const int fa = f - NP;
#pragma unroll 4
    for (int v = 0; v < NV; ++v) {
      float e = aggB[v * NA + a];
      float p = e * aggB[v * NA + fa];
      mx = fmaxf(mx, p); sm += p;
    }
  }
  float* outc = collapsed + ((size_t)b * NA + a) * TWOF;
  outc[f]         = mx;
  outc[NFEAT + f] = sm * (1.0f / NV);
}

// ------------- Kernel 3: Mtot[b,a,nf] = Wout_a + collapsed @ Wout_e ----------
__global__ __launch_bounds__(128) void
k_mtot(const float* __restrict__ collapsed, const float* __restrict__ Wout,
       float* __restrict__ Mtot)
{
  const int ba = blockIdx.x;                     // b*8 + a
  const int a  = ba & 7;
  const int nf = threadIdx.x;                    // 0..127
  const float* col = collapsed + (size_t)ba * TWOF;
  const float* We  = Wout + (size_t)(NP + a * TWOF) * NFO;    // expanded rows
  float s = Wout[(size_t)(NP + NA * TWOF + a) * NFO + nf];    // agg row 1216+a
#pragma unroll 8
  for (int g = 0; g < TWOF; ++g)
    s += col[g] * We[g * NFO + nf];
  Mtot[(size_t)ba * NFO + nf] = s;
}

// --- Kernel 4: out = tanh(X@Wout_v + agg@Mtot[b] + bout) ---------------------
// 256 blocks x 16 row-tiles; 8 waves own the 8 NF column tiles with persistent
// B fragments; X/agg tiles staged through LDS once per tile.
__global__ __launch_bounds__(256) void
k_out(const float* __restrict__ X, const float* __restrict__ agg,
      const float* __restrict__ Mtot, const float* __restrict__ Wout,
      const float* __restrict__ bout, float* __restrict__ out)
{
  __shared__ float xs[16 * XS_LD];
  __shared__ float as[16 * AS_LD];
  const int tid  = threadIdx.x;
  const int lane = tid & 31;
  const int wave = tid >> 5;                  // 0..7 -> NF column tile
  const int m    = lane & 15;
  const int kh   = (lane >> 4) * 2;
  const int n    = lane & 15;
  const int ncol = wave * 16 + n;
  const int blockrow = blockIdx.x * 256;      // 256 rows, all in one batch
  const int b = blockrow >> 11;               // blockrow / NV
  const float* Mb = Mtot + (size_t)b * NA * NFO;

  // persistent B fragments: 16 for X@Wout_v, 2 for agg@Mtot
  v2f bfr[16], bm[2];
#pragma unroll
  for (int kk = 0; kk < 16; ++kk) {
    const int kb = kk * 4 + kh;
    bfr[kk].x = Wout[kb * NFO + ncol];
    bfr[kk].y = Wout[(kb + 1) * NFO + ncol];
  }
#pragma unroll
  for (int kk = 0; kk < 2; ++kk) {
    const int kb = kk * 4 + kh;
    bm[kk].x = Mb[kb * NFO + ncol];
    bm[kk].y = Mb[(kb + 1) * NFO + ncol];
  }
  const float bb = bout[ncol];

#pragma unroll 1
  for (int t = 0; t < 16; ++t) {
    const int rowb = blockrow + t * 16;
    { // stage 16x64 X tile: exactly one float4 chunk per thread
      const int r = tid >> 4, c = (tid & 15) * 4;
      *(float4*)(xs + r * XS_LD + c) =
          *(const float4*)(X + (size_t)(rowb + r) * NP + c);
    }
    if (tid < 128) { // stage 16x8 agg tile
      const int r = tid >> 3, c = tid & 7;
      as[r * AS_LD + c] = agg[(size_t)(rowb + r) * NA + c];
    }
    __syncthreads();

    v8f acc = {0.f,0.f,0.f,0.f,0.f,0.f,0.f,0.f};
#pragma unroll
    for (int kk = 0; kk < 16; ++kk) {          // X @ Wout[0:64, :]
      const int kb = kk * 4 + kh;
      v2f av; av.x = xs[m * XS_LD + kb]; av.y = xs[m * XS_LD + kb + 1];
      acc = wmma_f32_k4(av, bfr[kk], acc);
    }
#pragma unroll
    for (int kk = 0; kk < 2; ++kk) {           // agg @ (M[b] + Wout_a)
      const int kb = kk * 4 + kh;
      v2f av; av.x = as[m * AS_LD + kb]; av.y = as[m * AS_LD + kb + 1];
      acc = wmma_f32_k4(av, bm[kk], acc);
    }

    const int mrow = rowb + (lane >> 4) * 8;
#pragma unroll
    for (int r = 0; r < 8; ++r)
      out[(size_t)(mrow + r) * NFO + ncol] = tanhf(acc[r] + bb);
    __syncthreads();
  }
}

extern "C" void kernel_launch(void* const* d_in, const int* in_sizes, int n_in,
                              void* d_out, int out_size, void* d_ws, size_t ws_size,
                              hipStream_t stream)
{
  const float* X    = (const float*)d_in[0];
  const float* W1   = (const float*)d_in[1];
  const float* b1   = (const float*)d_in[2];
  const float* W2   = (const float*)d_in[3];
  const float* b2   = (const float*)d_in[4];
  const float* Wout = (const float*)d_in[5];
  const float* bout = (const float*)d_in[6];
  float* out = (float*)d_out;

  float* ws        = (float*)d_ws;
  float* vi        = ws;                                    // 32*2048*64 f32
  float* agg       = vi  + (size_t)BATCH * NV * NP;         // 32*2048*8
  float* collapsed = agg + (size_t)BATCH * NV * NA;         // 32*8*144
  float* Mtot      = collapsed + (size_t)BATCH * NA * TWOF; // 32*8*128

  const int nBlocks = BATCH * NV / 256;                     // 256

  hipLaunchKernelGGL(k_feats,    dim3(nBlocks),    dim3(160), 0, stream,
                     X, W1, b1, W2, b2, vi, agg);
  hipLaunchKernelGGL(k_collapse, dim3(NA, BATCH),  dim3(96),  0, stream,
                     vi, agg, collapsed);
  hipLaunchKernelGGL(k_mtot,     dim3(BATCH * NA), dim3(128), 0, stream,
                     collapsed, Wout, Mtot);
  hipLaunchKernelGGL(k_out,      dim3(nBlocks),    dim3(256), 0, stream,
                     X, agg, Mtot, Wout, bout, out);
}